// NativeSparseAttention_8607114461311
// MI455X (gfx1250) — compile-verified
//
#include <hip/hip_runtime.h>
#include <math.h>
#include <stdint.h>

typedef float v2f __attribute__((ext_vector_type(2)));
typedef float v8f __attribute__((ext_vector_type(8)));

constexpr int Bc   = 2;
constexpr int Tc   = 2048;
constexpr int Cc   = 1024;
constexpr int NHc  = 16;
constexpr int HSc  = 64;
constexpr int Lc   = 32;
constexpr int STc  = 16;
constexpr int WINc = 256;
constexpr int NBc  = (Tc - Lc) / STc + 1;   // 127
constexpr int Mrows = Bc * Tc;              // 4096

__device__ __forceinline__ float gelu_exact(float x) {
  return 0.5f * x * (1.0f + erff(x * 0.70710678118654752440f));
}

__device__ __forceinline__ float apply_act(float x, int act) {
  if (act == 1) return gelu_exact(x);
  if (act == 2) return 1.0f / (1.0f + expf(-x));
  return x;
}

// ---------------------------------------------------------------------------
// C[M x N] = act(A[M x K] @ W[N x K]^T + bias[N])
// Block = 4 waves (128 threads) covering a 64(M) x 64(N) macro-tile.
// Wave w computes rows [tm + 16w, tm + 16w + 16) x 64 cols via
// V_WMMA_F32_16X16X4_F32. All 4 waves issue identical W addresses (same tn),
// so the WGP cache turns the W stream into hits for 3 of 4 waves.
// A fragment (16x4 f32): lane l holds A[l&15][k + 2*(l>>4) .. +1]
// B fragment (4x16 f32): lane l holds B[2*(l>>4) .. +1][l&15] = W[n][k..], n=l&15
// C/D (16x16 f32, 8 VGPRs): vgpr r -> row r + 8*(l>>4), col l&15
// ---------------------------------------------------------------------------
__global__ __launch_bounds__(128)
void gemm_wmma_f32(const float* __restrict__ A, const float* __restrict__ W,
                   const float* __restrict__ bias, float* __restrict__ Cout,
                   int Mdim, int Ndim, int Kdim, int act)
{
  (void)Mdim;
  const int lane = threadIdx.x & 31;
  const int wave = threadIdx.x >> 5;
  const int half = lane >> 4;
  const int l16  = lane & 15;
  const int tm = (blockIdx.y << 6) + (wave << 4);
  const int tn = blockIdx.x << 6;

  const float* a  = A + (size_t)(tm + l16)      * Kdim + (half << 1);
  const float* w0 = W + (size_t)(tn +  0 + l16) * Kdim + (half << 1);
  const float* w1 = W + (size_t)(tn + 16 + l16) * Kdim + (half << 1);
  const float* w2 = W + (size_t)(tn + 32 + l16) * Kdim + (half << 1);
  const float* w3 = W + (size_t)(tn + 48 + l16) * Kdim + (half << 1);

  v8f acc0 = {}; v8f acc1 = {}; v8f acc2 = {}; v8f acc3 = {};

  #pragma unroll 8
  for (int k = 0; k < Kdim; k += 4) {
    if ((k & 127) == 0) {          // CDNA5 global_prefetch_b8: stream next chunk
      __builtin_prefetch(a  + k + 128, 0, 0);
      __builtin_prefetch(w0 + k + 128, 0, 0);
      __builtin_prefetch(w1 + k + 128, 0, 0);
      __builtin_prefetch(w2 + k + 128, 0, 0);
      __builtin_prefetch(w3 + k + 128, 0, 0);
    }
    v2f av = *(const v2f*)(a  + k);
    v2f b0 = *(const v2f*)(w0 + k);
    v2f b1 = *(const v2f*)(w1 + k);
    v2f b2 = *(const v2f*)(w2 + k);
    v2f b3 = *(const v2f*)(w3 + k);
    acc0 = __builtin_amdgcn_wmma_f32_16x16x4_f32(false, av, false, b0, (short)0, acc0, false, false);
    acc1 = __builtin_amdgcn_wmma_f32_16x16x4_f32(false, av, false, b1, (short)0, acc1, false, false);
    acc2 = __builtin_amdgcn_wmma_f32_16x16x4_f32(false, av, false, b2, (short)0, acc2, false, false);
    acc3 = __builtin_amdgcn_wmma_f32_16x16x4_f32(false, av, false, b3, (short)0, acc3, false, false);
  }

  const int col = tn + l16;
  #pragma unroll
  for (int r = 0; r < 8; ++r) {
    const int row = tm + r + (half << 3);
    float* crow = Cout + (size_t)row * Ndim;
    crow[col]      = apply_act(acc0[r] + bias[col],      act);
    crow[col + 16] = apply_act(acc1[r] + bias[col + 16], act);
    crow[col + 32] = apply_act(acc2[r] + bias[col + 32], act);
    crow[col + 48] = apply_act(acc3[r] + bias[col + 48], act);
  }
}

// ---------------------------------------------------------------------------
// Compression MLP: for each (b,h,nb) block, 64 threads (one per head-dim hs):
//   xu[l]  = kv[b, nb*ST+l, part*C + h*HS + hs]
//   xpe    = xu + wpe.T ; h = gelu(xpe @ fc1W^T + fc1b)
//   out    = h @ fc2W^T + fc2b ; res = xu @ rpW^T + rpb
//   kcomp[b,h,nb,hs] = clip(out + res, -3, 3)
// fc1W (128x32 = 16KB) staged into LDS with CDNA5 async DMA
// (global_load_async_to_lds_b128, ASYNCcnt), bypassing VGPRs.
// ---------------------------------------------------------------------------
__global__ __launch_bounds__(64)
void comp_mlp_kernel(const float* __restrict__ kvbuf, int part,
                     const float* __restrict__ fc1W, const float* __restrict__ fc1b,
                     const float* __restrict__ fc2W, const float* __restrict__ fc2b,
                     const float* __restrict__ rpW,  const float* __restrict__ rpb,
                     const float* __restrict__ wpe,  float* __restrict__ outc)
{
  __shared__ __align__(16) float sW[4 * Lc * Lc];   // 4096 floats = 16 KB
  const int tid = threadIdx.x;

  // Async global -> LDS copy: each of the 64 lanes moves 256B as 16x B128 ops.
  {
    const char* gsrc = (const char*)fc1W + tid * 256;
    // generic LDS pointer carries the wave-relative LDS offset in its low 32 bits
    const uint32_t lbase = (uint32_t)(uintptr_t)(&sW[0]) + (uint32_t)(tid * 256);
    #pragma unroll
    for (int it = 0; it < 16; ++it) {
      uint32_t la = lbase + it * 16;
      const char* ga = gsrc + it * 16;
      asm volatile("global_load_async_to_lds_b128 %0, %1, off"
                   :: "v"(la), "v"(ga) : "memory");
    }
    asm volatile("s_wait_asynccnt 0x0" ::: "memory");
  }
  __syncthreads();

  const int nb = blockIdx.x % NBc;
  const int bh = blockIdx.x / NBc;
  const int h  = bh % NHc;
  const int b  = bh / NHc;
  const int hs = tid;

  const float* base = kvbuf + ((size_t)(b * Tc + nb * STc)) * (2 * Cc)
                      + (size_t)part * Cc + h * HSc + hs;
  float xu[Lc], xpe[Lc];
  #pragma unroll
  for (int l = 0; l < Lc; ++l) {
    float v = base[(size_t)l * (2 * Cc)];
    xu[l]  = v;
    xpe[l] = v + wpe[l * HSc + hs];
  }

  float out = fc2b[0];
  for (int j = 0; j < 4 * Lc; ++j) {
    float hj = fc1b[j];
    #pragma unroll
    for (int l = 0; l < Lc; ++l) hj = fmaf(xpe[l], sW[j * Lc + l], hj);
    out = fmaf(gelu_exact(hj), fc2W[j], out);
  }
  float res = rpb[0];
  #pragma unroll
  for (int l = 0; l < Lc; ++l) res = fmaf(xu[l], rpW[l], res);

  float r = fminf(3.0f, fmaxf(-3.0f, out + res));
  outc[((size_t)(b * NHc + h) * NBc + nb) * HSc + hs] = r;
}

// ---------------------------------------------------------------------------
// Local sliding-window attention: 1 block (64 threads) per (b,h,query i).
// Keys j in [max(0,i-WIN+1), i]. q staged in LDS, scores in LDS, tree reductions.
// ---------------------------------------------------------------------------
__global__ __launch_bounds__(64)
void local_attn_kernel(const float* __restrict__ qkv, float* __restrict__ y_local)
{
  __shared__ float qs[HSc];
  __shared__ float sc[WINc];
  __shared__ float red[64];
  const int t  = threadIdx.x;
  const int i  = blockIdx.x % Tc;
  const int bh = blockIdx.x / Tc;
  const int h  = bh % NHc;
  const int b  = bh / NHc;
  const float scale = 0.125f;   // 1/sqrt(64)

  const float* qp = qkv + ((size_t)(b * Tc + i)) * (3 * Cc) + h * HSc;
  qs[t] = qp[t];
  __syncthreads();

  const int jlo = (i >= WINc) ? (i - WINc + 1) : 0;
  const int nk  = i - jlo + 1;

  float lmax = -3.4e38f;
  for (int jj = t; jj < nk; jj += 64) {
    const float* kp = qkv + ((size_t)(b * Tc + jlo + jj)) * (3 * Cc) + Cc + h * HSc;
    float s = 0.0f;
    #pragma unroll
    for (int d = 0; d < HSc; ++d) s = fmaf(qs[d], kp[d], s);
    s *= scale;
    sc[jj] = s;
    lmax = fmaxf(lmax, s);
  }
  red[t] = lmax; __syncthreads();
  for (int s = 32; s > 0; s >>= 1) { if (t < s) red[t] = fmaxf(red[t], red[t + s]); __syncthreads(); }
  const float gmax = red[0];
  __syncthreads();

  float lsum = 0.0f;
  for (int jj = t; jj < nk; jj += 64) {
    float e = expf(sc[jj] - gmax);
    sc[jj] = e;
    lsum += e;
  }
  red[t] = lsum; __syncthreads();
  for (int s = 32; s > 0; s >>= 1) { if (t < s) red[t] += red[t + s]; __syncthreads(); }
  const float inv = 1.0f / red[0];

  float acc = 0.0f;
  for (int jj = 0; jj < nk; ++jj) {
    const float* vp = qkv + ((size_t)(b * Tc + jlo + jj)) * (3 * Cc) + 2 * Cc + h * HSc;
    acc = fmaf(sc[jj], vp[t], acc);
  }
  y_local[((size_t)(b * Tc + i)) * Cc + h * HSc + t] = acc * inv;
}

// ---------------------------------------------------------------------------
// Compressed-block attention: keys are NB=127 compressed blocks, mask nb*ST < i.
// i == 0 -> all masked -> softmax NaN -> nan_to_num -> zeros.
// ---------------------------------------------------------------------------
__global__ __launch_bounds__(64)
void comp_attn_kernel(const float* __restrict__ qkv,
                      const float* __restrict__ kc, const float* __restrict__ vc,
                      float* __restrict__ y_comp)
{
  __shared__ float qs[HSc];
  __shared__ float sc[128];
  __shared__ float red[64];
  const int t  = threadIdx.x;
  const int i  = blockIdx.x % Tc;
  const int bh = blockIdx.x / Tc;
  const int h  = bh % NHc;
  const int b  = bh / NHc;
  const float scale = 0.125f;

  int nallow = 0;
  if (i > 0) { nallow = ((i - 1) >> 4) + 1; if (nallow > NBc) nallow = NBc; }
  if (nallow == 0) {
    y_comp[((size_t)(b * Tc + i)) * Cc + h * HSc + t] = 0.0f;
    return;
  }

  const float* qp = qkv + ((size_t)(b * Tc + i)) * (3 * Cc) + h * HSc;
  qs[t] = qp[t];
  __syncthreads();

  const float* kbase = kc + (size_t)(b * NHc + h) * NBc * HSc;
  const float* vbase = vc + (size_t)(b * NHc + h) * NBc * HSc;

  float lmax = -3.4e38f;
  for (int nb = t; nb < nallow; nb += 64) {
    const float* kp = kbase + (size_t)nb * HSc;
    float s = 0.0f;
    #pragma unroll
    for (int d = 0; d < HSc; ++d) s = fmaf(qs[d], kp[d], s);
    s *= scale;
    sc[nb] = s;
    lmax = fmaxf(lmax, s);
  }
  red[t] = lmax; __syncthreads();
  for (int s = 32; s > 0; s >>= 1) { if (t < s) red[t] = fmaxf(red[t], red[t + s]); __syncthreads(); }
  const float gmax = red[0];
  __syncthreads();

  float lsum = 0.0f;
  for (int nb = t; nb < nallow; nb += 64) {
    float e = expf(sc[nb] - gmax);
    sc[nb] = e;
    lsum += e;
  }
  red[t] = lsum; __syncthreads();
  for (int s = 32; s > 0; s >>= 1) { if (t < s) red[t] += red[t + s]; __syncthreads(); }
  const float inv = 1.0f / red[0];

  float acc = 0.0f;
  for (int nb = 0; nb < nallow; ++nb)
    acc = fmaf(sc[nb], vbase[(size_t)nb * HSc + t], acc);
  y_comp[((size_t)(b * Tc + i)) * Cc + h * HSc + t] = acc * inv;
}

// y = gate*y_local + (1-gate)*y_comp
__global__ __launch_bounds__(256)
void mix_kernel(const float* __restrict__ gate, const float* __restrict__ yl,
                const float* __restrict__ yc, float* __restrict__ ym, int n)
{
  int idx = blockIdx.x * 256 + threadIdx.x;
  if (idx < n) {
    float g = gate[idx];
    ym[idx] = g * yl[idx] + (1.0f - g) * yc[idx];
  }
}

extern "C" void kernel_launch(void* const* d_in, const int* in_sizes, int n_in,
                              void* d_out, int out_size, void* d_ws, size_t ws_size,
                              hipStream_t stream)
{
  (void)in_sizes; (void)n_in; (void)out_size; (void)ws_size;
  const float* x      = (const float*)d_in[0];
  const float* Wqkv   = (const float*)d_in[1];
  const float* bqkv   = (const float*)d_in[2];
  const float* Wcomp  = (const float*)d_in[3];
  const float* bcomp  = (const float*)d_in[4];
  const float* Wproj  = (const float*)d_in[5];
  const float* bproj  = (const float*)d_in[6];
  const float* k_fc1W = (const float*)d_in[7];
  const float* k_fc1b = (const float*)d_in[8];
  const float* k_fc2W = (const float*)d_in[9];
  const float* k_fc2b = (const float*)d_in[10];
  const float* k_rpW  = (const float*)d_in[11];
  const float* k_rpb  = (const float*)d_in[12];
  const float* k_wpe  = (const float*)d_in[13];
  const float* v_fc1W = (const float*)d_in[14];
  const float* v_fc1b = (const float*)d_in[15];
  const float* v_fc2W = (const float*)d_in[16];
  const float* v_fc2b = (const float*)d_in[17];
  const float* v_rpW  = (const float*)d_in[18];
  const float* v_rpb  = (const float*)d_in[19];
  const float* v_wpe  = (const float*)d_in[20];
  const float* Wg1    = (const float*)d_in[21];
  const float* bg1    = (const float*)d_in[22];
  const float* Wg2    = (const float*)d_in[23];
  const float* bg2    = (const float*)d_in[24];

  float* ws = (float*)d_ws;
  size_t off = 0;
  float* qkv  = ws + off; off += (size_t)Mrows * 3 * Cc;          // 12.58M
  float* kv   = ws + off; off += (size_t)Mrows * 2 * Cc;          //  8.39M
  float* kcmp = ws + off; off += (size_t)Bc * NHc * NBc * HSc;    //  0.26M
  float* vcmp = ws + off; off += (size_t)Bc * NHc * NBc * HSc;    //  0.26M
  float* yloc = ws + off; off += (size_t)Mrows * Cc;              //  4.19M
  float* ycmp = ws + off; off += (size_t)Mrows * Cc;              //  4.19M
  float* g1   = ws + off; off += (size_t)Mrows * 4 * Cc;          // 16.78M
  float* gate = ws + off; off += (size_t)Mrows * Cc;              //  4.19M
  float* ymix = ws + off; off += (size_t)Mrows * Cc;              //  4.19M

  const dim3 blk128(128);

  // 1) qkv = x @ Wqkv^T + bqkv                      (4096x1024 @ 1024x3072)
  gemm_wmma_f32<<<dim3(3 * Cc / 64, Mrows / 64), blk128, 0, stream>>>(
      x, Wqkv, bqkv, qkv, Mrows, 3 * Cc, Cc, 0);
  // 2) kv = x @ Wcomp^T + bcomp                     (N=2048)
  gemm_wmma_f32<<<dim3(2 * Cc / 64, Mrows / 64), blk128, 0, stream>>>(
      x, Wcomp, bcomp, kv, Mrows, 2 * Cc, Cc, 0);
  // 3) g1 = gelu(x @ Wg1^T + bg1)                   (N=4096)
  gemm_wmma_f32<<<dim3(4 * Cc / 64, Mrows / 64), blk128, 0, stream>>>(
      x, Wg1, bg1, g1, Mrows, 4 * Cc, Cc, 1);
  // 4) gate = sigmoid(g1 @ Wg2^T + bg2)             (K=4096, N=1024)
  gemm_wmma_f32<<<dim3(Cc / 64, Mrows / 64), blk128, 0, stream>>>(
      g1, Wg2, bg2, gate, Mrows, Cc, 4 * Cc, 2);
  // 5) compression MLPs (k and v)
  comp_mlp_kernel<<<Bc * NHc * NBc, 64, 0, stream>>>(
      kv, 0, k_fc1W, k_fc1b, k_fc2W, k_fc2b, k_rpW, k_rpb, k_wpe, kcmp);
  comp_mlp_kernel<<<Bc * NHc * NBc, 64, 0, stream>>>(
      kv, 1, v_fc1W, v_fc1b, v_fc2W, v_fc2b, v_rpW, v_rpb, v_wpe, vcmp);
  // 6) local sliding-window attention
  local_attn_kernel<<<Bc * NHc * Tc, 64, 0, stream>>>(qkv, yloc);
  // 7) compressed-block attention
  comp_attn_kernel<<<Bc * NHc * Tc, 64, 0, stream>>>(qkv, kcmp, vcmp, ycmp);
  // 8) gated mix
  const int n = Mrows * Cc;
  mix_kernel<<<(n + 255) / 256, 256, 0, stream>>>(gate, yloc, ycmp, ymix, n);
  // 9) out = ymix @ Wproj^T + bproj
  gemm_wmma_f32<<<dim3(Cc / 64, Mrows / 64), blk128, 0, stream>>>(
      ymix, Wproj, bproj, (float*)d_out, Mrows, Cc, Cc, 0);
}